// UKIELoss_59588376264841
// MI455X (gfx1250) — compile-verified
//
#include <hip/hip_runtime.h>
#include <hip/hip_bf16.h>

typedef __attribute__((ext_vector_type(2))) float v2f;
typedef __attribute__((ext_vector_type(8))) float v8f;

#define B_SZ   4096
#define C_CLS  100
#define C_PAD  112          // ceil(100/16)*16
#define C_TILES 7
#define D_DIM  512
#define D_TILES 32          // 512/16
#define KSPLIT 8
#define IMG_N  (4096*3*64*64)
#define EPSV   1e-4f

// workspace layout (float offsets)
#define OFF_CE  0
#define OFF_SQ  1
#define OFF_MX  2
#define OFF_CNT 16
#define OFF_SX  128
#define OFF_SY  (OFF_SX + C_PAD*D_DIM)
#define OFF_DX  (OFF_SY + C_PAD*D_DIM)
#define OFF_DY  (OFF_DX + C_PAD)
#define WS_FLOATS (OFF_DY + C_PAD)

__global__ void init_ws_kernel(float* ws) {
    int i = blockIdx.x * blockDim.x + threadIdx.x;
    if (i < WS_FLOATS) ws[i] = 0.0f;
}

// one wave (32 lanes) per batch row: logsumexp over 100 logits + class counts
__global__ void ce_count_kernel(const float* __restrict__ logits,
                                const int* __restrict__ label,
                                float* __restrict__ ws) {
    int wave = (blockIdx.x * blockDim.x + threadIdx.x) >> 5;
    int lane = threadIdx.x & 31;
    if (wave >= B_SZ) return;
    const float* row = logits + (size_t)wave * C_CLS;
    float m = -INFINITY;
    for (int j = lane; j < C_CLS; j += 32) m = fmaxf(m, row[j]);
    for (int off = 16; off > 0; off >>= 1) m = fmaxf(m, __shfl_xor(m, off, 32));
    float s = 0.0f;
    for (int j = lane; j < C_CLS; j += 32) s += expf(row[j] - m);
    for (int off = 16; off > 0; off >>= 1) s += __shfl_xor(s, off, 32);
    if (lane == 0) {
        int lab = label[wave];
        float lse = m + logf(s);
        atomicAdd(&ws[OFF_CE], lse - row[lab]);
        atomicAdd(&ws[OFF_CNT + lab], 1.0f);
    }
}

// streaming 402MB reduction: sum((rec-img)^2) and max(img)
__global__ void mse_kernel(const float* __restrict__ rec,
                           const float* __restrict__ img,
                           float* __restrict__ ws) {
    size_t i = (size_t)blockIdx.x * blockDim.x + threadIdx.x;
    size_t stride = (size_t)gridDim.x * blockDim.x;
    const float4* r4 = (const float4*)rec;
    const float4* i4 = (const float4*)img;
    const size_t n4 = IMG_N / 4;
    float acc = 0.0f, mx = 0.0f;
    for (; i < n4; i += stride) {
        __builtin_prefetch(&r4[i + 8 * stride], 0, 0);   // global_prefetch_b8
        __builtin_prefetch(&i4[i + 8 * stride], 0, 0);
        float4 a = r4[i];
        float4 b = i4[i];
        float d0 = a.x - b.x, d1 = a.y - b.y, d2 = a.z - b.z, d3 = a.w - b.w;
        acc = fmaf(d0, d0, acc); acc = fmaf(d1, d1, acc);
        acc = fmaf(d2, d2, acc); acc = fmaf(d3, d3, acc);
        mx = fmaxf(mx, fmaxf(fmaxf(b.x, b.y), fmaxf(b.z, b.w)));
    }
    for (int off = 16; off > 0; off >>= 1) {
        acc += __shfl_xor(acc, off, 32);
        mx = fmaxf(mx, __shfl_xor(mx, off, 32));
    }
    __shared__ float sacc[8], smx[8];
    int lane = threadIdx.x & 31, w = threadIdx.x >> 5;
    if (lane == 0) { sacc[w] = acc; smx[w] = mx; }
    __syncthreads();
    if (threadIdx.x == 0) {
        float ta = 0.0f, tm = 0.0f;
        int nw = blockDim.x >> 5;
        for (int k = 0; k < nw; k++) { ta += sacc[k]; tm = fmaxf(tm, smx[k]); }
        atomicAdd(&ws[OFF_SQ], ta);
        atomicMax((unsigned int*)&ws[OFF_MX], __float_as_uint(tm));  // img >= 0
    }
}

// Segment sum as GEMM: sums[C,D] = OneHot[C,B] @ X[B,D], exact f32 WMMA.
// One wave per block. A-matrix 16x4 f32 layout: lanes 0-15 hold M=lane,
// VGPR0=K(2*half), VGPR1=K(2*half+1).  C/D 16x16 f32: VGPR j -> M=j+8*half, N=lane&15.
__global__ void __launch_bounds__(32)
segsum_wmma_kernel(const float* __restrict__ inv,
                   const float* __restrict__ var,
                   const int* __restrict__ label,
                   float* __restrict__ ws) {
    int idx = blockIdx.x;
    int kslice = idx & (KSPLIT - 1); idx >>= 3;
    int dtile  = idx & (D_TILES - 1); idx >>= 5;
    int ctile  = idx % C_TILES;
    int tensor = idx / C_TILES;

    const float* x = tensor ? var : inv;
    float* sums = ws + (tensor ? OFF_SY : OFF_SX);

    int lane = threadIdx.x;
    int half = lane >> 4;       // 0: K={0,1}, 1: K={2,3}
    int n    = lane & 15;
    int class_base = ctile * 16;
    int d_base     = dtile * 16;
    int k_begin = kslice * (B_SZ / KSPLIT);
    int k_end   = k_begin + (B_SZ / KSPLIT);

    v8f c = {};
    for (int k0 = k_begin; k0 < k_end; k0 += 4) {
        int kA = k0 + 2 * half;
        int l0 = label[kA];
        int l1 = label[kA + 1];
        v2f a, b;
        a[0] = (l0 == class_base + n) ? 1.0f : 0.0f;
        a[1] = (l1 == class_base + n) ? 1.0f : 0.0f;
        b[0] = x[(size_t)kA * D_DIM + d_base + n];
        b[1] = x[(size_t)(kA + 1) * D_DIM + d_base + n];
        // D = A(16x4 onehot) x B(4x16 x-chunk) + C   -> v_wmma_f32_16x16x4_f32
        c = __builtin_amdgcn_wmma_f32_16x16x4_f32(
                /*neg_a=*/false, a, /*neg_b=*/false, b,
                /*c_mod=*/(short)0, c, /*reuse_a=*/false, /*reuse_b=*/false);
    }
    for (int j = 0; j < 8; j++) {
        int row = class_base + j + 8 * half;
        if (row < C_CLS)
            atomicAdd(&sums[(size_t)row * D_DIM + d_base + n], c[j]);
    }
}

// wave per row: sum_d |x[b,d] - mu[label[b],d]| for both tensors
__global__ void dev_kernel(const float* __restrict__ inv,
                           const float* __restrict__ var,
                           const int* __restrict__ label,
                           float* __restrict__ ws) {
    int wave = (blockIdx.x * blockDim.x + threadIdx.x) >> 5;
    int lane = threadIdx.x & 31;
    if (wave >= B_SZ) return;
    int lab = label[wave];
    float rcp = 1.0f / fmaxf(ws[OFF_CNT + lab], 1.0f);
    const float* sx = ws + OFF_SX + (size_t)lab * D_DIM;
    const float* sy = ws + OFF_SY + (size_t)lab * D_DIM;
    const float* xr = inv + (size_t)wave * D_DIM;
    const float* yr = var + (size_t)wave * D_DIM;
    float dx = 0.0f, dy = 0.0f;
    for (int d = lane; d < D_DIM; d += 32) {
        dx += fabsf(xr[d] - sx[d] * rcp);
        dy += fabsf(yr[d] - sy[d] * rcp);
    }
    for (int off = 16; off > 0; off >>= 1) {
        dx += __shfl_xor(dx, off, 32);
        dy += __shfl_xor(dy, off, 32);
    }
    if (lane == 0) {
        atomicAdd(&ws[OFF_DX + lab], dx);
        atomicAdd(&ws[OFF_DY + lab], dy);
    }
}

__global__ void finalize_kernel(const float* __restrict__ ws, float* __restrict__ out) {
    if (threadIdx.x != 0 || blockIdx.x != 0) return;
    float cls_loss = ws[OFF_CE] / (float)B_SZ;
    float rec_loss = ws[OFF_SQ] / (float)IMG_N;
    float mx = __uint_as_float(((const unsigned int*)ws)[OFF_MX]);
    float psnr = 20.0f * log10f(mx / sqrtf(rec_loss));
    float inv_loss = 0.0f, var_loss = 0.0f, npres = 0.0f;
    for (int cc = 0; cc < C_CLS; cc++) {
        float cnt = ws[OFF_CNT + cc];
        if (cnt > 0.0f) {
            float safe = fmaxf(cnt, 1.0f);
            float sx = ws[OFF_DX + cc] / (safe * (float)D_DIM);
            float sy = ws[OFF_DY + cc] / (safe * (float)D_DIM);
            inv_loss += fmaxf(sx + EPSV, 0.0f);
            var_loss += fmaxf(1.0f - (sy + EPSV), 0.0f);
            npres += 1.0f;
        }
    }
    inv_loss /= npres;
    var_loss /= npres;
    float irep  = inv_loss + cls_loss;
    float total = rec_loss + inv_loss + var_loss + cls_loss;
    out[0] = cls_loss; out[1] = rec_loss; out[2] = psnr;
    out[3] = inv_loss; out[4] = var_loss; out[5] = irep; out[6] = total;
}

extern "C" void kernel_launch(void* const* d_in, const int* in_sizes, int n_in,
                              void* d_out, int out_size, void* d_ws, size_t ws_size,
                              hipStream_t stream) {
    const float* logits = (const float*)d_in[0];
    const float* rec    = (const float*)d_in[1];
    const float* inv    = (const float*)d_in[2];
    const float* var    = (const float*)d_in[3];
    const float* img    = (const float*)d_in[4];
    const int*   label  = (const int*)d_in[5];
    float* ws  = (float*)d_ws;
    float* out = (float*)d_out;

    init_ws_kernel<<<(WS_FLOATS + 255) / 256, 256, 0, stream>>>(ws);
    ce_count_kernel<<<(B_SZ * 32 + 255) / 256, 256, 0, stream>>>(logits, label, ws);
    mse_kernel<<<2048, 256, 0, stream>>>(rec, img, ws);
    segsum_wmma_kernel<<<2 * C_TILES * D_TILES * KSPLIT, 32, 0, stream>>>(inv, var, label, ws);
    dev_kernel<<<(B_SZ * 32 + 255) / 256, 256, 0, stream>>>(inv, var, label, ws);
    finalize_kernel<<<1, 32, 0, stream>>>(ws, out);
}